// GraphSAGE_79869211836878
// MI455X (gfx1250) — compile-verified
//
#include <hip/hip_runtime.h>

// GraphSAGE (3x SAGEConv + global mean pool + linear head) for MI455X/gfx1250.
//
// v2: atomic scatter replaced by CSR-by-destination gather (built once:
// degree histogram -> two-level LDS scan -> fill). Each layer's mean
// aggregation is pure coalesced loads (256B or 512B per edge) with zero feature
// atomics -- removes ~600M f32 L2 atomic RMWs, the previous roofline limiter.
// GEMMs run on V_WMMA_F32_16X16X4_F32, strip-per-wave (16x128 output per
// wave, 8 accumulators) so each A fragment is reused by 8 WMMAs in registers.
// Workspace requirement: ~170 MB.

constexpr int kNodes   = 100000;   // divisible by 16
constexpr int kHid     = 128;      // 8 column tiles of 16
constexpr int kGraphs  = 512;
constexpr int kClasses = 16;       // exactly one WMMA tile column

typedef __attribute__((ext_vector_type(2))) float v2f;
typedef __attribute__((ext_vector_type(8))) float v8f;

__device__ __forceinline__ v8f wmma4(v2f a, v2f b, v8f c) {
  // 8 args: (neg_a, A, neg_b, B, c_mod, C, reuse_a, reuse_b)
  return __builtin_amdgcn_wmma_f32_16x16x4_f32(false, a, false, b, (short)0, c,
                                               false, false);
}

// ---------------------------------------------------------------- utilities
__global__ void zero_f4(float4* __restrict__ p, int n4) {
  int i = blockIdx.x * blockDim.x + threadIdx.x;
  if (i < n4) p[i] = make_float4(0.f, 0.f, 0.f, 0.f);
}

__global__ void copy_i(const int* __restrict__ s, int* __restrict__ d, int n) {
  int i = blockIdx.x * blockDim.x + threadIdx.x;
  if (i < n) d[i] = s[i];
}

// ----------------------------------------------------------- CSR construction
__global__ void deg_count(const int* __restrict__ dst, int* __restrict__ deg,
                          int E) {
  int e = blockIdx.x * blockDim.x + threadIdx.x;
  if (e < E) atomicAdd(deg + dst[e], 1);
}

__global__ void block_sum(const int* __restrict__ deg, int* __restrict__ part,
                          int N) {
  __shared__ int sm[256];
  int i = blockIdx.x * 256 + threadIdx.x;
  sm[threadIdx.x] = (i < N) ? deg[i] : 0;
  __syncthreads();
  for (int off = 128; off > 0; off >>= 1) {
    if (threadIdx.x < off) sm[threadIdx.x] += sm[threadIdx.x + off];
    __syncthreads();
  }
  if (threadIdx.x == 0) part[blockIdx.x] = sm[0];
}

__global__ void scan_partials(int* __restrict__ part, int nb) {
  if (blockIdx.x == 0 && threadIdx.x == 0) {
    int run = 0;
    for (int i = 0; i < nb; ++i) { int v = part[i]; part[i] = run; run += v; }
  }
}

__global__ void block_scan(const int* __restrict__ deg, const int* __restrict__ part,
                           int* __restrict__ rowptr, int N, int E) {
  __shared__ int sm[256];
  int i = blockIdx.x * 256 + threadIdx.x;
  int v = (i < N) ? deg[i] : 0;
  sm[threadIdx.x] = v;
  __syncthreads();
  for (int off = 1; off < 256; off <<= 1) {   // Hillis-Steele inclusive scan
    int t = (threadIdx.x >= off) ? sm[threadIdx.x - off] : 0;
    __syncthreads();
    sm[threadIdx.x] += t;
    __syncthreads();
  }
  if (i < N) rowptr[i] = part[blockIdx.x] + sm[threadIdx.x] - v;  // exclusive
  if (i == 0) rowptr[N] = E;
}

__global__ void csr_fill(const int* __restrict__ src, const int* __restrict__ dst,
                         int* __restrict__ cursor, int* __restrict__ csr, int E) {
  int e = blockIdx.x * blockDim.x + threadIdx.x;
  if (e < E) {
    int pos = atomicAdd(cursor + dst[e], 1);
    csr[pos] = src[e];
  }
}

// ----------------------------------------------------- mean aggregation (CSR)
// agg[n,:] = mean over incoming edges of h[src,:]. One wave per node;
// coalesced 32*FP*4B row loads, prefetch of the next neighbor row.
template <int D>
__global__ void gather_mean(const float* __restrict__ h, const int* __restrict__ rowptr,
                            const int* __restrict__ csr, float* __restrict__ agg) {
  constexpr int FP = D / 32;   // floats per lane (2 or 4)
  int wave = (blockIdx.x * blockDim.x + threadIdx.x) >> 5;
  int lane = threadIdx.x & 31;
  if (wave >= kNodes) return;
  int beg = rowptr[wave];
  int end = rowptr[wave + 1];
  float a0 = 0.f, a1 = 0.f, a2 = 0.f, a3 = 0.f;
  for (int j = beg; j < end; ++j) {
    int s = csr[j];
    if (j + 1 < end) __builtin_prefetch(h + (long)csr[j + 1] * D, 0, 1);
    const float* hs = h + (long)s * D + lane * FP;
    if (FP == 4) {
      float4 v = *(const float4*)hs;
      a0 += v.x; a1 += v.y; a2 += v.z; a3 += v.w;
    } else {
      float2 v = *(const float2*)hs;
      a0 += v.x; a1 += v.y;
    }
  }
  float inv = 1.0f / fmaxf((float)(end - beg), 1.0f);   // mean (deg 0 -> 0)
  float* ag = agg + (long)wave * D + lane * FP;
  if (FP == 4) *(float4*)ag = make_float4(a0 * inv, a1 * inv, a2 * inv, a3 * inv);
  else         *(float2*)ag = make_float2(a0 * inv, a1 * inv);
}

// ---------------------------------------------------- fused SAGEConv GEMM
// out = act( agg @ Wl + bl + xin @ Wr ), out: [kNodes, kHid].
// One wave per 16-row strip: 8 accumulator tiles, A fragment reused 8x.
template <int K, bool RELU>
__global__ void sage_gemm(const float* __restrict__ agg, const float* __restrict__ xin,
                          const float* __restrict__ Wl, const float* __restrict__ bl,
                          const float* __restrict__ Wr, float* __restrict__ out) {
  int wave = (blockIdx.x * blockDim.x + threadIdx.x) >> 5;
  if (wave >= kNodes / 16) return;       // wave-uniform: EXEC stays all-1s
  int lane = threadIdx.x & 31;
  int half = lane >> 4;                  // K pair {0,1} vs {2,3}
  int l15  = lane & 15;
  int row  = wave * 16 + l15;            // A-matrix row held by this lane

  const float* arow = agg + (long)row * K;
  const float* xrow = xin + (long)row * K;

  v8f zero = {};
  v8f acc[8];
#pragma unroll
  for (int t = 0; t < 8; ++t) acc[t] = zero;

#pragma unroll 4
  for (int k0 = 0; k0 < K; k0 += 4) {
    int kk = k0 + half * 2;
    float2 av = *(const float2*)(arow + kk);
    v2f a; a.x = av.x; a.y = av.y;
    const float* w0 = Wl + (long)kk * kHid + l15;
#pragma unroll
    for (int t = 0; t < 8; ++t) {
      v2f b; b.x = w0[t * 16]; b.y = w0[kHid + t * 16];
      acc[t] = wmma4(a, b, acc[t]);
    }
  }
#pragma unroll 4
  for (int k0 = 0; k0 < K; k0 += 4) {
    int kk = k0 + half * 2;
    float2 xv = *(const float2*)(xrow + kk);
    v2f a; a.x = xv.x; a.y = xv.y;
    const float* w0 = Wr + (long)kk * kHid + l15;
#pragma unroll
    for (int t = 0; t < 8; ++t) {
      v2f b; b.x = w0[t * 16]; b.y = w0[kHid + t * 16];
      acc[t] = wmma4(a, b, acc[t]);
    }
  }

  // C/D layout: VGPR v <-> M = v + 8*half, lane l15 <-> N
  float* obase = out + (long)(wave * 16 + half * 8) * kHid + l15;
#pragma unroll
  for (int t = 0; t < 8; ++t) {
    float bias = bl[t * 16 + l15];
#pragma unroll
    for (int v = 0; v < 8; ++v) {
      float val = acc[t][v] + bias;
      if (RELU) val = fmaxf(val, 0.0f);
      obase[(long)v * kHid + t * 16] = val;
    }
  }
}

// ------------------------------------------------------------ graph pooling
__global__ void graph_count(const int* __restrict__ batch, float* __restrict__ gcnt,
                            int N) {
  int n = blockIdx.x * blockDim.x + threadIdx.x;
  if (n < N) atomicAdd(gcnt + batch[n], 1.0f);
}

__global__ void pool_scatter(const float* __restrict__ h, const int* __restrict__ batch,
                             float* __restrict__ gsum, int N) {
  int i = blockIdx.x * blockDim.x + threadIdx.x;   // N*32 threads
  if (i >= N * 32) return;
  int n = i >> 5;
  int f = (i & 31) << 2;
  int g = batch[n];
  float4 v = *(const float4*)(h + (long)n * kHid + f);
  float* p = gsum + (long)g * kHid + f;
  atomicAdd(p + 0, v.x);
  atomicAdd(p + 1, v.y);
  atomicAdd(p + 2, v.z);
  atomicAdd(p + 3, v.w);
}

// out = (gsum/max(gcnt,1)) @ W_out + b_out ;  [512,128] @ [128,16]
__global__ void out_gemm(const float* __restrict__ gsum, const float* __restrict__ gcnt,
                         const float* __restrict__ W, const float* __restrict__ b,
                         float* __restrict__ out) {
  int wave = (blockIdx.x * blockDim.x + threadIdx.x) >> 5;   // 0..31 row tiles
  int lane = threadIdx.x & 31;
  int half = lane >> 4;
  int l15  = lane & 15;
  int row  = wave * 16 + l15;
  int col  = l15;

  float inv = 1.0f / fmaxf(gcnt[row], 1.0f);
  const float* grow = gsum + (long)row * kHid;

  v8f acc = {};
#pragma unroll
  for (int k0 = 0; k0 < kHid; k0 += 4) {
    int kk = k0 + half * 2;
    float2 gv = *(const float2*)(grow + kk);
    v2f a;  a.x = gv.x * inv; a.y = gv.y * inv;
    v2f bb; bb.x = W[kk * kClasses + col]; bb.y = W[(kk + 1) * kClasses + col];
    acc = wmma4(a, bb, acc);
  }

  float bias = b[col];
  float* orow = out + (long)(wave * 16 + half * 8) * kClasses + col;
#pragma unroll
  for (int v = 0; v < 8; ++v) orow[v * kClasses] = acc[v] + bias;
}

// -------------------------------------------------------------------- launch
extern "C" void kernel_launch(void* const* d_in, const int* in_sizes, int n_in,
                              void* d_out, int out_size, void* d_ws, size_t ws_size,
                              hipStream_t stream) {
  (void)n_in; (void)out_size; (void)ws_size;

  const float* x     = (const float*)d_in[0];
  const int*   edge  = (const int*)d_in[1];
  const int*   batch = (const int*)d_in[2];
  const float* Wl1 = (const float*)d_in[3];
  const float* bl1 = (const float*)d_in[4];
  const float* Wr1 = (const float*)d_in[5];
  const float* Wl2 = (const float*)d_in[6];
  const float* bl2 = (const float*)d_in[7];
  const float* Wr2 = (const float*)d_in[8];
  const float* Wl3 = (const float*)d_in[9];
  const float* bl3 = (const float*)d_in[10];
  const float* Wr3 = (const float*)d_in[11];
  const float* Wout = (const float*)d_in[12];
  const float* bout = (const float*)d_in[13];

  const int E = in_sizes[1] / 2;          // edge_index is [2, E], row-major
  const int* src = edge;
  const int* dst = edge + E;

  // workspace carve-up (256B-aligned sections)
  auto al = [](size_t x) { return (x + 255) & ~(size_t)255; };
  char* w = (char*)d_ws;
  int* deg    = (int*)w;  w += al((size_t)kNodes * 4);
  int* rowptr = (int*)w;  w += al((size_t)(kNodes + 1) * 4);
  int* cursor = (int*)w;  w += al((size_t)kNodes * 4);
  int* part   = (int*)w;  w += al(512 * 4);
  int* csr    = (int*)w;  w += al((size_t)E * 4);
  float* agg  = (float*)w;  w += (size_t)kNodes * kHid * 4;
  float* hA   = (float*)w;  w += (size_t)kNodes * kHid * 4;
  float* hB   = (float*)w;  w += (size_t)kNodes * kHid * 4;
  float* gsum = (float*)w;  w += (size_t)kGraphs * kHid * 4;
  float* gcnt = (float*)w;

  const int T = 256;
  auto nb = [](long n, int t) { return (int)((n + t - 1) / t); };
  const int scan_blocks = (kNodes + 255) / 256;              // 391
  const int gemm_blocks = nb((long)(kNodes / 16) * 32, T);   // 6250 waves
  const int gath_blocks = nb((long)kNodes * 32, T);          // 1 wave/node

  // ---- CSR by destination (built once, reused by all 3 layers)
  zero_f4<<<nb(kNodes / 4, T), T, 0, stream>>>((float4*)deg, kNodes / 4);
  deg_count<<<nb(E, T), T, 0, stream>>>(dst, deg, E);
  block_sum<<<scan_blocks, 256, 0, stream>>>(deg, part, kNodes);
  scan_partials<<<1, 1, 0, stream>>>(part, scan_blocks);
  block_scan<<<scan_blocks, 256, 0, stream>>>(deg, part, rowptr, kNodes, E);
  copy_i<<<nb(kNodes, T), T, 0, stream>>>(rowptr, cursor, kNodes);
  csr_fill<<<nb(E, T), T, 0, stream>>>(src, dst, cursor, csr, E);

  // ---- layer 1: K = 64
  gather_mean<64><<<gath_blocks, T, 0, stream>>>(x, rowptr, csr, agg);
  sage_gemm<64, true><<<gemm_blocks, T, 0, stream>>>(agg, x, Wl1, bl1, Wr1, hA);

  // ---- layer 2: K = 128
  gather_mean<128><<<gath_blocks, T, 0, stream>>>(hA, rowptr, csr, agg);
  sage_gemm<128, true><<<gemm_blocks, T, 0, stream>>>(agg, hA, Wl2, bl2, Wr2, hB);

  // ---- layer 3: K = 128, no relu
  gather_mean<128><<<gath_blocks, T, 0, stream>>>(hB, rowptr, csr, agg);
  sage_gemm<128, false><<<gemm_blocks, T, 0, stream>>>(agg, hB, Wl3, bl3, Wr3, hA);

  // ---- global mean pool + linear head
  zero_f4<<<nb(kGraphs * kHid / 4, T), T, 0, stream>>>((float4*)gsum,
                                                       kGraphs * kHid / 4);
  zero_f4<<<nb(kGraphs / 4, T), T, 0, stream>>>((float4*)gcnt, kGraphs / 4);
  graph_count<<<nb(kNodes, T), T, 0, stream>>>(batch, gcnt, kNodes);
  pool_scatter<<<nb((long)kNodes * 32, T), T, 0, stream>>>(hA, batch, gsum, kNodes);
  out_gemm<<<(kGraphs / 16) * 32 / T, T, 0, stream>>>(gsum, gcnt, Wout, bout,
                                                      (float*)d_out);
}